// DIoULoss_65051574665319
// MI455X (gfx1250) — compile-verified
//
#include <hip/hip_runtime.h>

// Per-lane vector types for V_WMMA_F32_16X16X4_F32 on gfx1250 (wave32):
//   A: 16x4 f32  -> 64 elems / 32 lanes = 2 VGPRs -> float2
//   B: 4x16 f32  -> 64 elems / 32 lanes = 2 VGPRs -> float2
//   C/D: 16x16 f32 -> 256 / 32 = 8 VGPRs -> float8
typedef __attribute__((ext_vector_type(2))) float v2f;
typedef __attribute__((ext_vector_type(8))) float v8f;

#define DIOU_EPS 1e-7f

// Full-wave (32 lane) f32 sum using the matrix unit.
// A (16x4 f32, ISA 7.12.2): lane L<16 holds A[L][0],A[L][1]; lane L>=16 holds
// A[L-16][2],A[L-16][3]. Put partial in vgpr0, 0 in vgpr1:
//   A[m][0]=acc[m], A[m][2]=acc[m+16], A[m][1]=A[m][3]=0.
// With B = ones, D[m][n] = acc[m]+acc[m+16]. D vgpr j = row j (lanes 0-15) /
// row j+8 (lanes 16-31); sum(c[0..7]) gives half-wave sums, one xor-16
// exchange finishes. Exact f32.
__device__ __forceinline__ float wave_reduce_f32(float acc) {
  v2f a;
  a.x = acc;
  a.y = 0.0f;
  v2f b;
  b.x = 1.0f;
  b.y = 1.0f;
  v8f c = {};
  // 8 args: (neg_a, A, neg_b, B, c_mod, C, reuse_a, reuse_b)
  c = __builtin_amdgcn_wmma_f32_16x16x4_f32(false, a, false, b, (short)0, c,
                                            false, false);
  float s = c[0] + c[1] + c[2] + c[3] + c[4] + c[5] + c[6] + c[7];
  s += __shfl_xor(s, 16, 32);  // combine low-half / high-half row sums
  return s;                    // full 32-lane sum, identical in every lane
}

// DIoU loss for one xyxy box pair. Divides implemented with v_rcp_f32
// (TRANS op, co-executes with VALU on CDNA5) instead of the ~10-instruction
// IEEE div sequence; 1-ulp rcp is far inside tolerance for a mean loss.
__device__ __forceinline__ float diou_loss(float4 p, float4 q) {
  const float a1 = (p.z - p.x) * (p.w - p.y);
  const float a2 = (q.z - q.x) * (q.w - q.y);

  const float wx = fmaxf(fminf(p.z, q.z) - fmaxf(p.x, q.x), 0.0f);
  const float wy = fmaxf(fminf(p.w, q.w) - fmaxf(p.y, q.y), 0.0f);
  const float inter = wx * wy;
  const float uni = a1 + a2 - inter;

  const float wcx = fmaxf(p.z, q.z) - fminf(p.x, q.x);
  const float wcy = fmaxf(p.w, q.w) - fminf(p.y, q.y);
  const float area_c = wcx * wcy;

  const float iou = inter * __builtin_amdgcn_rcpf(uni);
  const float giou = iou - (area_c - uni) * __builtin_amdgcn_rcpf(area_c);

  // centers are (min+max)/2 -> delta = 0.5*((p.x+p.z)-(q.x+q.z))
  const float dx = (p.x + p.z) - (q.x + q.z);
  const float dy = (p.y + p.w) - (q.y + q.w);
  const float d2 = 0.25f * (dx * dx + dy * dy);
  const float diag2 = wcx * wcx + wcy * wcy;

  const float diou = giou - d2 * __builtin_amdgcn_rcpf(diag2 + DIOU_EPS);
  return 1.0f - diou;
}

// Stage 1: streaming masked DIoU accumulate, one f32 partial per block.
// 4 pairs per thread per step: 8x global_load_b128 + 1x b32 mask load,
// branchless masked FMA accumulate.
__global__ void __launch_bounds__(256) diou_partial_kernel(
    const float4* __restrict__ boxes_in, const float4* __restrict__ boxes_tg,
    const unsigned char* __restrict__ mask, float* __restrict__ partials,
    int n) {
  const int tid = blockIdx.x * blockDim.x + threadIdx.x;
  const int stride = gridDim.x * blockDim.x;
  const int ngrp = n >> 2;  // groups of 4 pairs

  float acc = 0.0f;
  for (int g = tid; g < ngrp; g += stride) {
    const int base = g << 2;
    const uchar4 m = ((const uchar4*)mask)[g];

    const float4 p0 = boxes_in[base + 0];
    const float4 p1 = boxes_in[base + 1];
    const float4 p2 = boxes_in[base + 2];
    const float4 p3 = boxes_in[base + 3];
    const float4 q0 = boxes_tg[base + 0];
    const float4 q1 = boxes_tg[base + 1];
    const float4 q2 = boxes_tg[base + 2];
    const float4 q3 = boxes_tg[base + 3];

    acc = fmaf((float)m.x, diou_loss(p0, q0), acc);
    acc = fmaf((float)m.y, diou_loss(p1, q1), acc);
    acc = fmaf((float)m.z, diou_loss(p2, q2), acc);
    acc = fmaf((float)m.w, diou_loss(p3, q3), acc);
  }

  // Scalar tail (n % 4 != 0); reconverges before the wave reduce.
  for (int i = (ngrp << 2) + tid; i < n; i += stride) {
    const float mf = mask[i] ? 1.0f : 0.0f;
    acc = fmaf(mf, diou_loss(boxes_in[i], boxes_tg[i]), acc);
  }

  // Wave reduce via matrix unit (EXEC all-1s: loops have reconverged).
  const float wsum = wave_reduce_f32(acc);

  __shared__ float smem[8];  // 256 threads = 8 wave32s
  const int lane = threadIdx.x & 31;
  const int wid = threadIdx.x >> 5;
  if (lane == 0) smem[wid] = wsum;
  __syncthreads();

  if (threadIdx.x == 0) {
    float t = 0.0f;
#pragma unroll
    for (int j = 0; j < 8; ++j) t += smem[j];
    partials[blockIdx.x] = t;  // fixed order -> deterministic
  }
}

// Stage 2: single block folds per-block partials; deterministic order.
__global__ void __launch_bounds__(256) diou_final_kernel(
    const float* __restrict__ partials, int nparts,
    const int* __restrict__ num_boxes, float* __restrict__ out) {
  float acc = 0.0f;
  for (int i = threadIdx.x; i < nparts; i += blockDim.x) acc += partials[i];

  const float wsum = wave_reduce_f32(acc);

  __shared__ float smem[8];
  const int lane = threadIdx.x & 31;
  const int wid = threadIdx.x >> 5;
  if (lane == 0) smem[wid] = wsum;
  __syncthreads();

  if (threadIdx.x == 0) {
    float t = 0.0f;
#pragma unroll
    for (int j = 0; j < 8; ++j) t += smem[j];
    out[0] = t / (float)num_boxes[0];
  }
}

extern "C" void kernel_launch(void* const* d_in, const int* in_sizes, int n_in,
                              void* d_out, int out_size, void* d_ws,
                              size_t ws_size, hipStream_t stream) {
  (void)n_in;
  (void)out_size;
  (void)ws_size;

  const float4* boxes_in = (const float4*)d_in[0];            // (B,Q,4) f32
  const float4* boxes_tg = (const float4*)d_in[1];            // (B,Q,4) f32
  const unsigned char* mask = (const unsigned char*)d_in[2];  // (B,Q) bool
  const int* num_boxes = (const int*)d_in[3];                 // scalar int

  float* out = (float*)d_out;
  float* partials = (float*)d_ws;  // BLOCKS floats (8 KB)

  const int n = in_sizes[0] / 4;  // number of box pairs (B*Q)

  constexpr int THREADS = 256;  // 8 wave32s per block
  constexpr int BLOCKS = 2048;  // ~16K waves in flight -> saturate 23.3 TB/s

  diou_partial_kernel<<<BLOCKS, THREADS, 0, stream>>>(boxes_in, boxes_tg, mask,
                                                      partials, n);
  diou_final_kernel<<<1, THREADS, 0, stream>>>(partials, BLOCKS, num_boxes,
                                               out);
}